// Aggregator_32959579030024
// MI455X (gfx1250) — compile-verified
//
#include <hip/hip_runtime.h>
#include <hip/hip_bf16.h>

// Aggregator: out[n,d] = curr[n,d] + sum_k alpha[n,k] * msg[n,k,d]
// N=100000, K=32, D=128, fp32. AI ~0.47 FLOP/B -> pure HBM stream (~75us floor
// at 23.3 TB/s). msg is streamed through the CDNA5 async global->LDS path,
// double-buffered per wave with s_wait_asynccnt pipelining. wave32-exact:
// one alpha per k (scalar path), one float4 of d per lane.

#define K_DIM 32
#define D_DIM 128
#define WPB 4                    // waves per block (wave32)
#define BLOCK_THREADS (WPB * 32)
#define ROW_BYTES (K_DIM * D_DIM * 4)   // 16 KiB per msg row
#define HALF_BYTES (ROW_BYTES / 2)      // 8 KiB half-row (16 k-slices)

typedef int v4i __attribute__((vector_size(4 * sizeof(int))));
typedef __attribute__((address_space(1))) v4i g_v4i_t;
typedef __attribute__((address_space(3))) v4i lds_v4i_t;
typedef float f32x4 __attribute__((ext_vector_type(4)));

__device__ __forceinline__ void async_fence() {
  asm volatile("" ::: "memory");
}

__device__ __forceinline__ void wait_async_0() {
#if __has_builtin(__builtin_amdgcn_s_wait_asynccnt)
  __builtin_amdgcn_s_wait_asynccnt(0);
#else
  asm volatile("s_wait_asynccnt 0x0" ::: "memory");
#endif
}

__device__ __forceinline__ void wait_async_16() {
#if __has_builtin(__builtin_amdgcn_s_wait_asynccnt)
  __builtin_amdgcn_s_wait_asynccnt(16);
#else
  asm volatile("s_wait_asynccnt 0x10" ::: "memory");
#endif
}

// Copy one 8 KiB half-row (16 k-slices x 512B) global -> LDS, async.
// One base address per lane; the 16 chunks use the instruction's immediate
// offset (ISA: INST_OFFSET is added to BOTH the global and the LDS address),
// so no per-chunk VALU address math.
__device__ __forceinline__ void async_half(const float* __restrict__ msg_row,
                                           int half, char* lds_buf, int lane) {
  const char* g = (const char*)msg_row + half * HALF_BYTES + lane * 16;
  char* l = lds_buf + lane * 16;
#define ASYNC_CHUNK(OFF)                                                     \
  __builtin_amdgcn_global_load_async_to_lds_b128((g_v4i_t*)g, (lds_v4i_t*)l, \
                                                 (OFF), 0)
  ASYNC_CHUNK(0);    ASYNC_CHUNK(512);  ASYNC_CHUNK(1024); ASYNC_CHUNK(1536);
  ASYNC_CHUNK(2048); ASYNC_CHUNK(2560); ASYNC_CHUNK(3072); ASYNC_CHUNK(3584);
  ASYNC_CHUNK(4096); ASYNC_CHUNK(4608); ASYNC_CHUNK(5120); ASYNC_CHUNK(5632);
  ASYNC_CHUNK(6144); ASYNC_CHUNK(6656); ASYNC_CHUNK(7168); ASYNC_CHUNK(7680);
#undef ASYNC_CHUNK
}

// Constant-index component select on an ext-vector (folds after unroll).
__device__ __forceinline__ float f4c(f32x4 v, int c) {
  switch (c & 3) {
    case 0: return v.x;
    case 1: return v.y;
    case 2: return v.z;
    default: return v.w;
  }
}

// Accumulate 16 k-slices (one half-row) from LDS into acc.
// av[] holds the 32 wave-uniform alphas (registers after full unroll).
__device__ __forceinline__ void compute_half(const char* lds_buf, int half,
                                             const f32x4 (&av)[8], f32x4& acc,
                                             int lane) {
  const f32x4* b = (const f32x4*)lds_buf;
#pragma unroll
  for (int j = 0; j < 16; ++j) {
    const int k = half * 16 + j;
    const float ak = f4c(av[k >> 2], k);     // wave-uniform scalar operand
    const f32x4 m = b[j * 32 + lane];        // ds_load_b128
    acc += m * ak;                           // 4x v_fmac_f32 (s-src alpha)
  }
}

extern "C" __global__ __launch_bounds__(BLOCK_THREADS)
void Aggregator_32959579030024_kernel(const float* __restrict__ curr,
                                      const float* __restrict__ alpha,
                                      const float* __restrict__ msg,
                                      float* __restrict__ out, int N) {
  // Per-wave double buffer: 2 x 8 KiB. 4 waves -> 64 KiB static LDS.
  __shared__ __align__(16) char smem[WPB * 2 * HALF_BYTES];

  const int lane = threadIdx.x & 31;
  const int wave = threadIdx.x >> 5;
  char* buf[2];
  buf[0] = smem + wave * (2 * HALF_BYTES);
  buf[1] = buf[0] + HALF_BYTES;

  const int stride = gridDim.x * WPB;
  int n = blockIdx.x * WPB + wave;
  if (n >= N) return;

  // Prologue: start streaming half 0 of the first row.
  async_half(msg + (size_t)n * (K_DIM * D_DIM), 0, buf[0], lane);

  int p = 0;
  while (true) {
    // n is wave-uniform; make it provably so -> scalar (SGPR) alpha loads.
    const int nu = __builtin_amdgcn_readfirstlane(n);
    const float* mrow = msg + (size_t)nu * (K_DIM * D_DIM);

    // Wave-uniform alpha row: 8 uniform b128 loads, overlapped with async.
    f32x4 av[8];
    const f32x4* arow = (const f32x4*)(alpha + (size_t)nu * K_DIM);
#pragma unroll
    for (int i = 0; i < 8; ++i) av[i] = arow[i];

    f32x4 acc = *((const f32x4*)(curr + (size_t)nu * D_DIM) + lane);

    // --- half 0: prefetch half 1, wait for half 0, consume half 0 ---
    async_half(mrow, 1, buf[p ^ 1], lane);
    wait_async_16();           // half 0 done (async loads complete in order)
    async_fence();
    compute_half(buf[p], 0, av, acc, lane);
    p ^= 1;

    // --- half 1: prefetch next row's half 0 (if any), wait, consume ---
    const int nn = n + stride;
    const bool more = (nn < N);
    if (more) {
      async_half(msg + (size_t)nn * (K_DIM * D_DIM), 0, buf[p ^ 1], lane);
      wait_async_16();
    } else {
      wait_async_0();
    }
    async_fence();
    compute_half(buf[p], 1, av, acc, lane);
    p ^= 1;

    // Streamed once, never re-read: nontemporal b128 store.
    __builtin_nontemporal_store(acc, (f32x4*)(out + (size_t)nu * D_DIM) + lane);

    if (!more) break;
    n = nn;
  }
}

extern "C" void kernel_launch(void* const* d_in, const int* in_sizes, int n_in,
                              void* d_out, int out_size, void* d_ws, size_t ws_size,
                              hipStream_t stream) {
  const float* curr  = (const float*)d_in[0];  // (N,1,D) f32
  const float* alpha = (const float*)d_in[1];  // (N,K,1) f32
  const float* msg   = (const float*)d_in[2];  // (N,K,D) f32
  float* out = (float*)d_out;                  // (N,D) f32

  const int N = in_sizes[0] / D_DIM;

  int blocks_needed = (N + WPB - 1) / WPB;
  int blocks = blocks_needed < 2048 ? blocks_needed : 2048;  // persistent waves
  Aggregator_32959579030024_kernel<<<blocks, BLOCK_THREADS, 0, stream>>>(
      curr, alpha, msg, out, N);
}